// GATLayer_17592186044977
// MI455X (gfx1250) — compile-verified
//
#include <hip/hip_runtime.h>
#include <math.h>

typedef __attribute__((ext_vector_type(2))) float v2f;
typedef __attribute__((ext_vector_type(4))) float f4;
typedef __attribute__((ext_vector_type(8))) float v8f;

#define HEADS 8
#define DK 16
#define DIM 128
#define NEG_SLOPE 0.2f

// padded LDS strides (in floats) to avoid bank conflicts
#define WTS 130   // W-transposed row stride: bank = (2*col + k) % 64, 8B aligned for b64
#define ASP 132   // A-tile row stride: bank = (4*row + k) % 64, 16B aligned for b128

// ---- monotonic float <-> uint key (for atomic segment-max) ----
__device__ __forceinline__ unsigned f2key(float f) {
    unsigned u = __float_as_uint(f);
    return (u & 0x80000000u) ? ~u : (u | 0x80000000u);
}
__device__ __forceinline__ float key2f(unsigned k) {
    unsigned u = (k & 0x80000000u) ? (k & 0x7FFFFFFFu) : ~k;
    return __uint_as_float(u);
}
__device__ __forceinline__ float leaky(float x) {
    return x > 0.f ? x : NEG_SLOPE * x;
}

// ---- init: t accumulator (=d_out), max keys (= -inf), denom (=0) ----
__global__ void k_init(float* __restrict__ t, unsigned* __restrict__ mkey,
                       float* __restrict__ denom, int n_nodes) {
    int i = blockIdx.x * blockDim.x + threadIdx.x;
    if (i < n_nodes * DIM) t[i] = 0.f;
    if (i < n_nodes * HEADS) {
        mkey[i]  = 0x007FFFFFu;   // f2key(-inf)
        denom[i] = 0.f;
    }
}

// ---- GEMM hp = h @ W via V_WMMA_F32_16X16X4_F32, plus per-node logits ----
// Block = 256 threads = 8 waves. Each wave computes one 16x128 tile of hp.
// LDS: WT (128 cols x 130-stride, transposed W), then 8 waves x (16 x 132) A tiles.
__global__ void k_gemm_logits(const float* __restrict__ h,
                              const float* __restrict__ W,
                              const float* __restrict__ att_l,
                              const float* __restrict__ att_r,
                              float* __restrict__ hp,
                              float* __restrict__ a_src,
                              float* __restrict__ a_dst,
                              int n_nodes) {
    extern __shared__ float smem[];
    float* WT = smem;                               // WT[col*WTS + k] = W[k][col]
    const int tid  = threadIdx.x;
    const int wave = tid >> 5;
    const int lane = tid & 31;
    float* As = smem + DIM * WTS + wave * (16 * ASP);  // this wave's 16-row tile

    // stage W transposed (coalesced float4 reads, scattered scalar LDS writes)
    {
        const f4* Wg = (const f4*)W;
#pragma unroll
        for (int i = 0; i < 16; ++i) {
            int sIdx = tid + 256 * i;      // 0..4095 float4 slots of W (row-major)
            int k    = sIdx >> 5;          // W row (K index)
            int c4   = sIdx & 31;          // float4 column group
            f4  v    = Wg[sIdx];
            WT[(4 * c4 + 0) * WTS + k] = v.x;
            WT[(4 * c4 + 1) * WTS + k] = v.y;
            WT[(4 * c4 + 2) * WTS + k] = v.z;
            WT[(4 * c4 + 3) * WTS + k] = v.w;
        }
    }
    // stage this wave's 16 rows of h (zero-pad out-of-range rows)
    const int row0 = blockIdx.x * 128 + wave * 16;
    {
#pragma unroll
        for (int i = 0; i < 16; ++i) {
            int s    = i * 32 + lane;   // 0..511 float4 slots
            int r    = s >> 5;          // row in tile
            int c4   = s & 31;          // float4 col
            int grow = row0 + r;
            f4  v    = {0.f, 0.f, 0.f, 0.f};
            if (grow < n_nodes) v = ((const f4*)h)[(size_t)grow * 32 + c4];
            *(f4*)(As + r * ASP + c4 * 4) = v;
        }
    }
    __syncthreads();

    const int half = lane >> 4;   // 0: K pair {0,1} / D rows 0-7; 1: K pair {2,3} / D rows 8-15
    const int m    = lane & 15;

    v8f acc[8];
#pragma unroll
    for (int c = 0; c < 8; ++c) acc[c] = v8f{0, 0, 0, 0, 0, 0, 0, 0};

    // K = 128 in steps of 4
    for (int s = 0; s < 32; ++s) {
        int k0 = 4 * s + half * 2;
        // A 16x4 fragment: lane<16 -> (M=m, K=k0,k0+1); lane>=16 -> (M=m, K=k0,k0+1 = 4s+2,4s+3)
        v2f a = *(const v2f*)(As + m * ASP + k0);
#pragma unroll
        for (int c = 0; c < 8; ++c) {
            // B 4x16 fragment: contiguous b64 from transposed W
            v2f b = *(const v2f*)(WT + (c * 16 + m) * WTS + k0);
            acc[c] = __builtin_amdgcn_wmma_f32_16x16x4_f32(
                false, a, false, b, (short)0, acc[c], false, false);
        }
    }

    // D layout: VGPR r, lanes 0-15 -> (M=r, N=lane); lanes 16-31 -> (M=8+r, N=lane-16)
    // write to global hp + back into LDS tile (for logits)
#pragma unroll
    for (int c = 0; c < 8; ++c) {
#pragma unroll
        for (int r = 0; r < 8; ++r) {
            int rr  = r + half * 8;
            int col = c * 16 + m;
            As[rr * ASP + col] = acc[c][r];
            int grow = row0 + rr;
            if (grow < n_nodes) hp[(size_t)grow * DIM + col] = acc[c][r];
        }
    }
    __syncthreads();

    // logits: 16 rows x 8 heads = 128 (row,head) pairs per wave
#pragma unroll
    for (int i = 0; i < 4; ++i) {
        int p    = i * 32 + lane;
        int r    = p >> 3;
        int hh   = p & 7;
        int grow = row0 + r;
        if (grow < n_nodes) {
            float dl = 0.f, dr = 0.f;
#pragma unroll
            for (int d = 0; d < DK; ++d) {
                float v = As[r * ASP + hh * DK + d];
                dl += v * att_l[hh * DK + d];
                dr += v * att_r[hh * DK + d];
            }
            a_src[(size_t)grow * HEADS + hh] = dl;
            a_dst[(size_t)grow * HEADS + hh] = dr;
        }
    }
}

// ---- edge pass 1: segment max of leaky_relu(a_src[src]+a_dst[dst]) over dst ----
__global__ void k_edge_max(const int* __restrict__ src, const int* __restrict__ dst,
                           const float* __restrict__ a_src, const float* __restrict__ a_dst,
                           unsigned* __restrict__ mkey, int n_edges) {
    int e = blockIdx.x * blockDim.x + threadIdx.x;
    if (e >= n_edges) return;
    int s = src[e], d = dst[e];
    const f4* ps = (const f4*)(a_src + (size_t)s * HEADS);
    const f4* pd = (const f4*)(a_dst + (size_t)d * HEADS);
#pragma unroll
    for (int q = 0; q < 2; ++q) {
        f4 vs = ps[q], vd = pd[q];
#pragma unroll
        for (int j = 0; j < 4; ++j) {
            float x = leaky(vs[j] + vd[j]);
            atomicMax(&mkey[(size_t)d * HEADS + q * 4 + j], f2key(x));
        }
    }
}

// ---- edge pass 2: denom[dst] += exp(e - m[dst]) ----
__global__ void k_edge_sum(const int* __restrict__ src, const int* __restrict__ dst,
                           const float* __restrict__ a_src, const float* __restrict__ a_dst,
                           const unsigned* __restrict__ mkey, float* __restrict__ denom,
                           int n_edges) {
    int e = blockIdx.x * blockDim.x + threadIdx.x;
    if (e >= n_edges) return;
    int s = src[e], d = dst[e];
    const f4* ps = (const f4*)(a_src + (size_t)s * HEADS);
    const f4* pd = (const f4*)(a_dst + (size_t)d * HEADS);
#pragma unroll
    for (int q = 0; q < 2; ++q) {
        f4 vs = ps[q], vd = pd[q];
#pragma unroll
        for (int j = 0; j < 4; ++j) {
            int   hh = q * 4 + j;
            float x  = leaky(vs[j] + vd[j]);
            float mm = key2f(mkey[(size_t)d * HEADS + hh]);
            atomicAdd(&denom[(size_t)d * HEADS + hh], __expf(x - mm));
        }
    }
}

// ---- edge pass 3: t[dst] += hp[src] * alpha ; one thread per (edge, head) ----
__global__ void k_aggregate(const int* __restrict__ src, const int* __restrict__ dst,
                            const float* __restrict__ a_src, const float* __restrict__ a_dst,
                            const unsigned* __restrict__ mkey, const float* __restrict__ denom,
                            const float* __restrict__ hp, float* __restrict__ t,
                            int n_edges) {
    int tid = blockIdx.x * blockDim.x + threadIdx.x;
    int e   = tid >> 3;
    int hh  = tid & 7;
    if (e >= n_edges) return;
    int s = src[e], d = dst[e];
    float x     = leaky(a_src[(size_t)s * HEADS + hh] + a_dst[(size_t)d * HEADS + hh]);
    float mm    = key2f(mkey[(size_t)d * HEADS + hh]);
    float den   = denom[(size_t)d * HEADS + hh];
    float alpha = __expf(x - mm) / fmaxf(den, 1e-16f);

    const f4* hv = (const f4*)(hp + (size_t)s * DIM + hh * DK);
    float*    tb = t + (size_t)d * DIM + hh * DK;
#pragma unroll
    for (int q = 0; q < 4; ++q) {
        f4 v = hv[q];
        atomicAdd(tb + q * 4 + 0, v.x * alpha);
        atomicAdd(tb + q * 4 + 1, v.y * alpha);
        atomicAdd(tb + q * 4 + 2, v.z * alpha);
        atomicAdd(tb + q * 4 + 3, v.w * alpha);
    }
}

// ---- final: out = t*sigmoid(skip) + h*(1-sigmoid(skip)), in place on d_out ----
__global__ void k_final(const float* __restrict__ h, const float* __restrict__ skip,
                        float* __restrict__ out, int n_nodes) {
    int i = blockIdx.x * blockDim.x + threadIdx.x;
    if (i >= n_nodes * DIM) return;
    float a = 1.f / (1.f + __expf(-skip[0]));
    out[i] = out[i] * a + h[i] * (1.f - a);
}

extern "C" void kernel_launch(void* const* d_in, const int* in_sizes, int n_in,
                              void* d_out, int out_size, void* d_ws, size_t ws_size,
                              hipStream_t stream) {
    const float* h     = (const float*)d_in[0];
    const float* W     = (const float*)d_in[1];
    const float* att_l = (const float*)d_in[2];
    const float* att_r = (const float*)d_in[3];
    const float* skip  = (const float*)d_in[4];
    const int*   src   = (const int*)d_in[5];
    const int*   dst   = (const int*)d_in[6];

    const int n_nodes = in_sizes[0] / DIM;
    const int n_edges = in_sizes[5];
    float*    out     = (float*)d_out;

    // workspace layout (floats): hp | a_src | a_dst | mkey(u32) | denom
    float*    hp    = (float*)d_ws;
    float*    a_src = hp + (size_t)n_nodes * DIM;
    float*    a_dst = a_src + (size_t)n_nodes * HEADS;
    unsigned* mkey  = (unsigned*)(a_dst + (size_t)n_nodes * HEADS);
    float*    denom = (float*)(mkey + (size_t)n_nodes * HEADS);

    const int BS = 256;

    k_init<<<(n_nodes * DIM + BS - 1) / BS, BS, 0, stream>>>(out, mkey, denom, n_nodes);

    const int    gemm_blocks = (n_nodes + 127) / 128;
    const size_t lds_bytes   = (DIM * WTS + 8 * 16 * ASP) * sizeof(float);  // ~131 KB
    k_gemm_logits<<<gemm_blocks, BS, lds_bytes, stream>>>(h, W, att_l, att_r,
                                                          hp, a_src, a_dst, n_nodes);

    k_edge_max<<<(n_edges + BS - 1) / BS, BS, 0, stream>>>(src, dst, a_src, a_dst,
                                                           mkey, n_edges);
    k_edge_sum<<<(n_edges + BS - 1) / BS, BS, 0, stream>>>(src, dst, a_src, a_dst,
                                                           mkey, denom, n_edges);
    k_aggregate<<<((size_t)n_edges * 8 + BS - 1) / BS, BS, 0, stream>>>(
        src, dst, a_src, a_dst, mkey, denom, hp, out, n_edges);

    k_final<<<(n_nodes * DIM + BS - 1) / BS, BS, 0, stream>>>(h, skip, out, n_nodes);
}